// MDL_67542655697718
// MI455X (gfx1250) — compile-verified
//
#include <hip/hip_runtime.h>
#include <hip/hip_bf16.h>
#include <stdint.h>

// ---------------------------------------------------------------------------
// MDL feature hashing + per-feature binning (gfx1250 / MI455X).
//
// Memory plan:
//   - keys/vals/outs stream from HBM (128 MB total -> ~5.5us floor at
//     23.3 TB/s), moved as b128 vectors (int4/float4) + global_prefetch of
//     the next grid-stride chunk to overlap with the dependent searches.
//   - hash_keys (400KB) / bin tables (~41MB) are L2-resident (192MB L2).
//   - A coarse sample of hash_keys (stride 64) is staged into LDS per block
//     via CDNA5 async global->LDS loads (ASYNCcnt + s_wait_asynccnt), so the
//     100K-entry lower_bound costs ~11 LDS steps + <=6 L2 steps per entry.
//   - Bin upper_bound (51 boundaries) only executes for hash hits; EXEC
//     masking under wave32 suppresses the loads for miss lanes.
//   - No WMMA: this is a zero-FLOP gather/search workload.
// ---------------------------------------------------------------------------

#define SAMPLE_STRIDE 64
#define MAX_SAMPLES   8192   // 32 KB LDS; supports n_feature up to 512K

struct MdlTables {
    const int*   hash_keys;
    const int*   hash_values;
    const int*   bin_ids;
    const float* bin_values;
    const int*   feature_offsets;
    int n_feature;
    int n_bin_p1;
    int mdl_size;
    int non_mdl_size;
    int n_samples;
    int use_coarse;
};

__device__ __forceinline__ void mdl_lookup_one(
    const MdlTables& t, const int* __restrict__ s_coarse,
    int key, float val, int& okey, float& oval)
{
    // ---- lower_bound(hash_keys, key): coarse LDS search, then fine L2 search
    int lo, hi;
    if (t.use_coarse) {
        int clo = 0, chi = t.n_samples;                 // upper_bound over samples
        while (clo < chi) {
            int mid = (clo + chi) >> 1;
            if (s_coarse[mid] <= key) clo = mid + 1; else chi = mid;
        }
        int jstar = clo - 1;                            // last sample <= key
        if (jstar < 0) { lo = 0; hi = 0; }              // key below table -> pos 0
        else {
            lo = jstar * SAMPLE_STRIDE;
            hi = min(lo + SAMPLE_STRIDE, t.n_feature);
        }
    } else {
        lo = 0; hi = t.n_feature;
    }
    while (lo < hi) {                                   // fine lower_bound
        int mid = (lo + hi) >> 1;
        if (t.hash_keys[mid] < key) lo = mid + 1; else hi = mid;
    }
    const int  idx   = min(lo, t.n_feature - 1);
    const bool found = (t.hash_keys[idx] == key);

    if (found) {
        // ---- MDL path: upper_bound into this feature's bin boundaries
        const int offset = t.feature_offsets[t.hash_values[idx]];
        int blo = 0, bhi = t.n_bin_p1;
        while (blo < bhi) {
            int mid = (blo + bhi) >> 1;
            if (t.bin_values[offset + mid] <= val) blo = mid + 1; else bhi = mid;
        }
        int bin_idx = min(max(blo - 1, 0), t.n_bin_p1 - 1);
        okey = t.bin_ids[offset + bin_idx];
        oval = 1.0f;
    } else {
        // ---- non-MDL path: remap into [mdl_size, output_size)
        int r = key % t.non_mdl_size;
        if (r < 0) r += t.non_mdl_size;                 // match jnp.mod semantics
        okey = r + t.mdl_size;
        oval = val;
    }
}

__global__ void MDL_67542655697718_kernel(
    const int*   __restrict__ keys,
    const float* __restrict__ vals,
    int*         __restrict__ out_keys,
    float*       __restrict__ out_vals,
    int n, MdlTables t)
{
    __shared__ int s_coarse[MAX_SAMPLES];
    const int tid = threadIdx.x;

    // ---- Stage coarse hash-key samples into LDS with async global->LDS copies.
    if (t.use_coarse) {
        for (int j = tid; j < t.n_samples; j += blockDim.x) {
            // Low 32 bits of a generic pointer to __shared__ are the LDS offset.
            uint32_t lds_off = (uint32_t)(size_t)(&s_coarse[j]);
            const int* gptr  = t.hash_keys + (size_t)j * SAMPLE_STRIDE;
            asm volatile("global_load_async_to_lds_b32 %0, %1, off"
                         :: "v"(lds_off), "v"(gptr) : "memory");
        }
        asm volatile("s_wait_asynccnt 0" ::: "memory");
    }
    __syncthreads();

    const int gtid    = blockIdx.x * blockDim.x + tid;
    const int gstride = gridDim.x * blockDim.x;
    const int nvec    = n >> 2;

    const int4*   keys4 = (const int4*)keys;
    const float4* vals4 = (const float4*)vals;
    int4*         ok4   = (int4*)out_keys;
    float4*       ov4   = (float4*)out_vals;

    // ---- Main vectorized stream: b128 loads/stores, 4 entries per iteration.
    for (int v = gtid; v < nvec; v += gstride) {
        // Prefetch next chunk of this thread's stream (global_prefetch_b8).
        if (v + gstride < nvec) {
            __builtin_prefetch((const void*)(keys4 + v + gstride), 0, 3);
            __builtin_prefetch((const void*)(vals4 + v + gstride), 0, 3);
        }
        const int4   k4 = keys4[v];
        const float4 f4 = vals4[v];
        int4 ok; float4 ov;
        mdl_lookup_one(t, s_coarse, k4.x, f4.x, ok.x, ov.x);
        mdl_lookup_one(t, s_coarse, k4.y, f4.y, ok.y, ov.y);
        mdl_lookup_one(t, s_coarse, k4.z, f4.z, ok.z, ov.z);
        mdl_lookup_one(t, s_coarse, k4.w, f4.w, ok.w, ov.w);
        ok4[v] = ok;
        ov4[v] = ov;
    }

    // ---- Scalar tail for n not divisible by 4.
    for (int i = (nvec << 2) + gtid; i < n; i += gstride) {
        int okey; float oval;
        mdl_lookup_one(t, s_coarse, keys[i], vals[i], okey, oval);
        out_keys[i] = okey;
        out_vals[i] = oval;
    }
}

extern "C" void kernel_launch(void* const* d_in, const int* in_sizes, int n_in,
                              void* d_out, int out_size, void* d_ws, size_t ws_size,
                              hipStream_t stream) {
    // setup_inputs order: ids, keys, vals, hash_keys, hash_values,
    //                     bin_ids, bin_values, feature_offsets
    const int*   keys = (const int*)  d_in[1];
    const float* vals = (const float*)d_in[2];

    MdlTables t;
    t.hash_keys       = (const int*)  d_in[3];
    t.hash_values     = (const int*)  d_in[4];
    t.bin_ids         = (const int*)  d_in[5];
    t.bin_values      = (const float*)d_in[6];
    t.feature_offsets = (const int*)  d_in[7];

    const int n   = in_sizes[1];
    t.n_feature   = in_sizes[3];
    t.mdl_size    = in_sizes[5];                       // n_feature * (n_bin+1)
    t.n_bin_p1    = t.mdl_size / t.n_feature;
    const int output_size = 1 << 24;                   // OUT_BITS = 24
    t.non_mdl_size = output_size - t.mdl_size;

    t.n_samples  = (t.n_feature + SAMPLE_STRIDE - 1) / SAMPLE_STRIDE;
    t.use_coarse = (t.n_samples <= MAX_SAMPLES) ? 1 : 0;
    if (!t.use_coarse) t.n_samples = 0;

    int*   out_keys = (int*)d_out;                     // first n: int32 keys
    float* out_vals = (float*)d_out + n;               // next n:  float32 vals

    const int block        = 256;                      // 8 wave32 waves
    const int vec_per_thr  = 4;                        // 16 entries per thread
    const int nvec         = n >> 2;
    int grid = (nvec + block * vec_per_thr - 1) / (block * vec_per_thr);
    if (grid < 1) grid = 1;

    MDL_67542655697718_kernel<<<grid, block, 0, stream>>>(
        keys, vals, out_keys, out_vals, n, t);
}